// ScaledDotProductAttention_74612171866395
// MI455X (gfx1250) — compile-verified
//
#include <hip/hip_runtime.h>

typedef float v2f __attribute__((ext_vector_type(2)));
typedef float v8f __attribute__((ext_vector_type(8)));

#define BATCH 4
#define HEADS 8
#define SEQ   2048
#define DIM   64
#define NT    (SEQ / 16)      // 128 key tiles
#define NW    4               // mask words per row-stream (128 tiles / 32)
#define ATT_SCALE 0.125f
#define NEGMIN (-3.402823466e38f)

__device__ __forceinline__ v8f wmma_f32(v2f a, v2f b, v8f c) {
  // D = A(16x4 f32) * B(4x16 f32) + C(16x16 f32)
  return __builtin_amdgcn_wmma_f32_16x16x4_f32(
      /*neg_a=*/false, a, /*neg_b=*/false, b,
      /*c_mod=*/(short)0, c, /*reuse_a=*/false, /*reuse_b=*/false);
}

__global__ __launch_bounds__(256)
void sdpa_fp32_wmma_kernel(const float* __restrict__ Q,
                           const float* __restrict__ K,
                           const float* __restrict__ V,
                           const int*   __restrict__ MASK,
                           float* __restrict__ Out,
                           float* __restrict__ Attn) {
  // Per-wave 16x16 P staging tile; stride 18 keeps float2 loads 8B-aligned.
  __shared__ float psh[8 * 16 * 18];

  const int wave = threadIdx.x >> 5;
  const int lane = threadIdx.x & 31;
  const int lo   = lane & 15;
  const int hi   = lane >> 4;

  const int bh = blockIdx.y;           // 0..31
  const int b  = bh / HEADS;
  const int q0 = blockIdx.x * 128 + wave * 16;

  const float* q  = Q + (size_t)bh * SEQ * DIM;
  const float* k  = K + (size_t)bh * SEQ * DIM;
  const float* v  = V + (size_t)bh * SEQ * DIM;
  const int*   mk = MASK + (size_t)b * SEQ * SEQ;   // mask is [B,1,S,S]
  float* outp  = Out  + (size_t)bh * SEQ * DIM;
  float* attnp = Attn + (size_t)bh * SEQ * SEQ;
  float* pw = psh + wave * (16 * 18);

  // ---- Load Q A-fragments (pre-scaled), kept resident for both passes.
  // A 16x4 layout: lane(0..15)=row M, VGPR0 holds K=0|2, VGPR1 holds K=1|3.
  v2f aq[16];
#pragma unroll
  for (int c = 0; c < 16; ++c) {
    const float* p = q + (size_t)(q0 + lo) * DIM + 4 * c + 2 * hi;
    aq[c].x = p[0] * ATT_SCALE;
    aq[c].y = p[1] * ATT_SCALE;
  }

  // Per-lane online-softmax stats and 1-bit/element mask cache.
  float lm[8], ll[8];
  unsigned mb[8][NW];
#pragma unroll
  for (int i = 0; i < 8; ++i) {
    lm[i] = NEGMIN;
    ll[i] = 0.0f;
  }

  // =================== PASS A: row max / sum-exp ===================
  // Chunked by mask word (w is a compile-time constant inside each copy)
  // so mb[][] never sees a dynamic index -> no movrel / exec-mask branches.
#pragma unroll
  for (int w = 0; w < NW; ++w) {
    unsigned mbw[8];
#pragma unroll
    for (int i = 0; i < 8; ++i) mbw[i] = 0u;

    for (int j2 = 0; j2 < 32; ++j2) {
      const int j = w * 32 + j2;

      v2f bk[16];
#pragma unroll
      for (int c = 0; c < 16; ++c) {
        const float* p = k + (size_t)(j * 16 + lo) * DIM + 4 * c + 2 * hi;
        bk[c].x = p[0];
        bk[c].y = p[1];
      }
      v8f c8 = {};
#pragma unroll
      for (int c = 0; c < 16; ++c) c8 = wmma_f32(aq[c], bk[c], c8);

#pragma unroll
      for (int i = 0; i < 8; ++i) {
        const int row = q0 + i + 8 * hi;
        const unsigned bit = (mk[(size_t)row * SEQ + j * 16 + lo] != 0) ? 1u : 0u;
        mbw[i] |= bit << j2;                 // branchless cndmask+shift+or
        const float s  = bit ? c8[i] : NEGMIN;
        // Single-exp online update: one of exp(lm-mn), exp(s-mn) is 1.
        const float old = lm[i];
        const float mn  = fmaxf(old, s);
        const float e   = __expf(fminf(old, s) - mn);
        const bool  up  = s > old;
        ll[i] = ll[i] * (up ? e : 1.0f) + (up ? 1.0f : e);
        lm[i] = mn;
      }
    }
#pragma unroll
    for (int i = 0; i < 8; ++i) mb[i][w] = mbw[i];
  }

  // Cross-lane combine within each 16-lane half (rows are split across 16 lanes).
  float mrow[8], ilrow[8];
#pragma unroll
  for (int i = 0; i < 8; ++i) {
    float m = lm[i], l = ll[i];
#pragma unroll
    for (int ofs = 1; ofs < 16; ofs <<= 1) {
      const float mo  = __shfl_xor(m, ofs, 32);
      const float lo2 = __shfl_xor(l, ofs, 32);
      const float mn  = fmaxf(m, mo);
      l = l * __expf(m - mn) + lo2 * __expf(mo - mn);
      m = mn;
    }
    mrow[i]  = m;
    ilrow[i] = 1.0f / l;   // l >= 1 whenever any element equals the row max
  }

  // =================== PASS B: probs + P@V ===================
  v8f acc[4];
  {
    const v8f z = {};
#pragma unroll
    for (int t = 0; t < 4; ++t) acc[t] = z;
  }

#pragma unroll
  for (int w = 0; w < NW; ++w) {
    for (int j2 = 0; j2 < 32; ++j2) {
      const int j = w * 32 + j2;

      // Recompute the 16x16 score tile.
      v2f bk[16];
#pragma unroll
      for (int c = 0; c < 16; ++c) {
        const float* p = k + (size_t)(j * 16 + lo) * DIM + 4 * c + 2 * hi;
        bk[c].x = p[0];
        bk[c].y = p[1];
      }
      v8f c8 = {};
#pragma unroll
      for (int c = 0; c < 16; ++c) c8 = wmma_f32(aq[c], bk[c], c8);

      // Normalized probabilities: write to global attn and stage in LDS.
#pragma unroll
      for (int i = 0; i < 8; ++i) {
        const unsigned keep = (mb[i][w] >> j2) & 1u;   // w is constant here
        const float s = keep ? c8[i] : NEGMIN;
        const float p = __expf(s - mrow[i]) * ilrow[i];
        attnp[(size_t)(q0 + i + 8 * hi) * SEQ + j * 16 + lo] = p;
        pw[(i + 8 * hi) * 18 + lo] = p;
      }
      // Same-wave LDS ops execute in order; each wave owns its slice -> no barrier.

      // Reload P as A-fragments (row = lo, key-chunk by hi).
      v2f pa[4];
#pragma unroll
      for (int kc = 0; kc < 4; ++kc) {
        const float* pp = pw + lo * 18 + 4 * kc + 2 * hi;
        pa[kc].x = pp[0];
        pa[kc].y = pp[1];
      }

      // acc[dt] += P(16x16) * V(16x64), K-dim = keys in chunks of 4.
#pragma unroll
      for (int kc = 0; kc < 4; ++kc) {
#pragma unroll
        for (int dt = 0; dt < 4; ++dt) {
          v2f bv;
          const float* pv0 =
              v + (size_t)(j * 16 + 4 * kc + 2 * hi) * DIM + dt * 16 + lo;
          bv.x = pv0[0];
          bv.y = pv0[DIM];
          acc[dt] = wmma_f32(pa[kc], bv, acc[dt]);
        }
      }
    }
  }

  // ---- Epilogue: write the 16x64 output tile.
#pragma unroll
  for (int dt = 0; dt < 4; ++dt) {
#pragma unroll
    for (int i = 0; i < 8; ++i) {
      outp[(size_t)(q0 + i + 8 * hi) * DIM + dt * 16 + lo] = acc[dt][i];
    }
  }
}

extern "C" void kernel_launch(void* const* d_in, const int* in_sizes, int n_in,
                              void* d_out, int out_size, void* d_ws, size_t ws_size,
                              hipStream_t stream) {
  (void)in_sizes; (void)n_in; (void)d_ws; (void)ws_size; (void)out_size;
  const float* q    = (const float*)d_in[0];
  const float* k    = (const float*)d_in[1];
  const float* v    = (const float*)d_in[2];
  const int*   mask = (const int*)d_in[3];

  float* out  = (float*)d_out;
  float* attn = out + (size_t)BATCH * HEADS * SEQ * DIM;  // tuple: (output, attn)

  dim3 grid(SEQ / 128, BATCH * HEADS);  // (16, 32)
  dim3 block(256);                      // 8 waves x 16 queries = 128 queries/block
  sdpa_fp32_wmma_kernel<<<grid, block, 0, stream>>>(q, k, v, mask, out, attn);
}